// NCINet_30648886624731
// MI455X (gfx1250) — compile-verified
//
#include <hip/hip_runtime.h>
#include <math.h>

// ---------------------------------------------------------------------------
// NCINet forward for MI455X (gfx1250), fp32 throughout via V_WMMA_F32_16X16X4_F32.
// alpha=2 Renyi entropy computed as 16 - log2(||K||_F^2)  (tr K == N == 256).
// ---------------------------------------------------------------------------

typedef __attribute__((ext_vector_type(2))) float v2f;
typedef __attribute__((ext_vector_type(8))) float v8f;

#define REG_    1.0f
#define BN_EPS_ 1e-5f

__device__ __forceinline__ v8f wmma4(v2f a, v2f b, v8f c) {
  // D = A(16x4,f32) * B(4x16,f32) + C(16x16,f32)
  return __builtin_amdgcn_wmma_f32_16x16x4_f32(false, a, false, b, (short)0, c,
                                               false, false);
}

// ---------------- mlp11: out = relu(in @ w1 + b1) @ w2 + b2  ([16384,64] x [64,64] x2)
__global__ __launch_bounds__(128) void k_mlp11(const float* __restrict__ in,
                                               const float* __restrict__ w1,
                                               const float* __restrict__ b1,
                                               const float* __restrict__ w2,
                                               const float* __restrict__ b2,
                                               float* __restrict__ out) {
  __shared__ float h[16][64];
  const int lane = threadIdx.x & 31, wave = threadIdx.x >> 5;
  const int m = lane & 15, half = lane >> 4;
  const int row0 = blockIdx.x * 16;
  const int col = wave * 16 + m;

  v8f acc = {};
  for (int k0 = 0; k0 < 64; k0 += 4) {
    const int ka = k0 + 2 * half;
    v2f a, bb;
    a.x  = in[(row0 + m) * 64 + ka];
    a.y  = in[(row0 + m) * 64 + ka + 1];
    bb.x = w1[ka * 64 + col];
    bb.y = w1[(ka + 1) * 64 + col];
    acc = wmma4(a, bb, acc);
  }
  const float bias1 = b1[col];
#pragma unroll
  for (int r = 0; r < 8; ++r)
    h[r + 8 * half][col] = fmaxf(acc[r] + bias1, 0.0f);
  __syncthreads();

  v8f acc2 = {};
  for (int k0 = 0; k0 < 64; k0 += 4) {
    const int ka = k0 + 2 * half;
    v2f a, bb;
    a.x  = h[m][ka];
    a.y  = h[m][ka + 1];
    bb.x = w2[ka * 64 + col];
    bb.y = w2[(ka + 1) * 64 + col];
    acc2 = wmma4(a, bb, acc2);
  }
  const float bias2 = b2[col];
#pragma unroll
  for (int r = 0; r < 8; ++r)
    out[(row0 + r + 8 * half) * 64 + col] = acc2[r] + bias2;
}

// ---------------- mlp12a: h1 = relu([out_x | out_y] @ w12_1 + b12_1)  ([16384,128]x[128,128])
__global__ __launch_bounds__(256) void k_mlp12a(const float* __restrict__ ox,
                                                const float* __restrict__ oy,
                                                const float* __restrict__ w,
                                                const float* __restrict__ b,
                                                float* __restrict__ h1) {
  const int lane = threadIdx.x & 31, wave = threadIdx.x >> 5;
  const int m = lane & 15, half = lane >> 4;
  const int row0 = blockIdx.x * 16;
  const int col = wave * 16 + m;

  v8f acc = {};
  for (int k0 = 0; k0 < 128; k0 += 4) {
    const float* src = (k0 < 64) ? ox : oy;
    const int kb = (k0 < 64) ? k0 : (k0 - 64);
    const int ka = kb + 2 * half;
    const int kw = k0 + 2 * half;
    v2f a, bb;
    a.x  = src[(row0 + m) * 64 + ka];
    a.y  = src[(row0 + m) * 64 + ka + 1];
    bb.x = w[kw * 128 + col];
    bb.y = w[(kw + 1) * 128 + col];
    acc = wmma4(a, bb, acc);
  }
  const float bias = b[col];
#pragma unroll
  for (int r = 0; r < 8; ++r)
    h1[(row0 + r + 8 * half) * 128 + col] = fmaxf(acc[r] + bias, 0.0f);
}

// ---------------- mlp12b: h2 = relu(h1 @ w12_2 + b12_2), mean over N -> [B,64]
__global__ __launch_bounds__(256) void k_mlp12b(const float* __restrict__ h1,
                                                const float* __restrict__ w,
                                                const float* __restrict__ b,
                                                float* __restrict__ meanc) {
  __shared__ float accum[64];
  const int bidx = blockIdx.x;
  const int lane = threadIdx.x & 31, wave = threadIdx.x >> 5;
  const int m = lane & 15, half = lane >> 4;
  if (threadIdx.x < 64) accum[threadIdx.x] = 0.0f;
  __syncthreads();

  for (int t = wave; t < 64; t += 8) {       // 16 row-tiles x 4 col-tiles
    const int ti = t >> 2, tj = t & 3;
    const int col = tj * 16 + m;
    v8f acc = {};
    for (int k0 = 0; k0 < 128; k0 += 4) {
      const int ka = k0 + 2 * half;
      v2f a, bb;
      a.x  = h1[(bidx * 256 + ti * 16 + m) * 128 + ka];
      a.y  = h1[(bidx * 256 + ti * 16 + m) * 128 + ka + 1];
      bb.x = w[ka * 64 + col];
      bb.y = w[(ka + 1) * 64 + col];
      acc = wmma4(a, bb, acc);
    }
    const float bias = b[col];
    float part = 0.0f;
#pragma unroll
    for (int r = 0; r < 8; ++r) part += fmaxf(acc[r] + bias, 0.0f);
    atomicAdd(&accum[col], part);            // ds_add_f32
  }
  __syncthreads();
  if (threadIdx.x < 64)
    meanc[bidx * 64 + threadIdx.x] = accum[threadIdx.x] * (1.0f / 256.0f);
}

// ---------------- head1: out_c = mean @ wc1 + bc1 ; prob1 = softmax
__global__ __launch_bounds__(64) void k_head1(const float* __restrict__ meanc,
                                              const float* __restrict__ wc1,
                                              const float* __restrict__ bc1,
                                              float* __restrict__ out_c,
                                              float* __restrict__ prob1) {
  const int b = threadIdx.x;
  float o[3] = {bc1[0], bc1[1], bc1[2]};
  for (int k = 0; k < 64; ++k) {
    const float v = meanc[b * 64 + k];
#pragma unroll
    for (int j = 0; j < 3; ++j) o[j] += v * wc1[k * 3 + j];
  }
  const float mx = fmaxf(o[0], fmaxf(o[1], o[2]));
  const float e0 = expf(o[0] - mx), e1 = expf(o[1] - mx), e2 = expf(o[2] - mx);
  const float inv = 1.0f / (e0 + e1 + e2);
#pragma unroll
  for (int j = 0; j < 3; ++j) out_c[b * 3 + j] = o[j];
  prob1[b * 3 + 0] = e0 * inv;
  prob1[b * 3 + 1] = e1 * inv;
  prob1[b * 3 + 2] = e2 * inv;
}

// ---------------- per-batch Frobenius normalize [N,D] slice
__global__ __launch_bounds__(256) void k_fronorm(const float* __restrict__ src,
                                                 float* __restrict__ dst) {
  __shared__ float red[256];
  const int b = blockIdx.x;
  const float* s = src + (size_t)b * (256 * 64);
  float acc = 0.0f;
  for (int i = threadIdx.x; i < 256 * 64; i += 256) {
    const float v = s[i];
    acc += v * v;
  }
  red[threadIdx.x] = acc;
  __syncthreads();
  for (int off = 128; off > 0; off >>= 1) {
    if (threadIdx.x < off) red[threadIdx.x] += red[threadIdx.x + off];
    __syncthreads();
  }
  const float inv = rsqrtf(red[0]);
  float* d = dst + (size_t)b * (256 * 64);
  for (int i = threadIdx.x; i < 256 * 64; i += 256) d[i] = s[i] * inv;
}

// ---------------- ridge regressor per batch: P4 = X (X^T X + ridge)^-1 X^T y
// resid = targ - P4 ; loss = ||resid||^2.  X = [1 | feat], 65x65 SPD Gauss-Jordan in LDS.
#define AUGW 130
__global__ __launch_bounds__(256) void k_regress(const float* __restrict__ feat,
                                                 const float* __restrict__ targ,
                                                 float* __restrict__ resid,
                                                 float* __restrict__ loss) {
  __shared__ float aug[65][AUGW];   // [P1 | I] -> [M | inv]
  __shared__ float A2[65][64];      // X^T y
  __shared__ float fac[65];
  __shared__ float red[256];
  const int b = blockIdx.x;
  const float* F = feat + (size_t)b * 256 * 64;
  const float* T = targ + (size_t)b * 256 * 64;

  // P1 = X^T X (+ ridge), augmented with identity
  for (int idx = threadIdx.x; idx < 65 * 65; idx += 256) {
    const int i = idx / 65, j = idx % 65;
    float s = 0.0f;
    for (int n = 0; n < 256; ++n) {
      const float xi = (i == 0) ? 1.0f : F[n * 64 + i - 1];
      const float xj = (j == 0) ? 1.0f : F[n * 64 + j - 1];
      s += xi * xj;
    }
    if (i == j && i > 0) s += REG_;
    aug[i][j] = s;
    aug[i][65 + j] = (i == j) ? 1.0f : 0.0f;
  }
  __syncthreads();

  // Gauss-Jordan, no pivoting (SPD, diag >= 1)
  for (int p = 0; p < 65; ++p) {
    const float invp = 1.0f / aug[p][p];
    __syncthreads();
    for (int c = threadIdx.x; c < 130; c += 256) aug[p][c] *= invp;
    __syncthreads();
    if (threadIdx.x < 65) fac[threadIdx.x] = aug[threadIdx.x][p];
    __syncthreads();
    for (int idx = threadIdx.x; idx < 65 * 130; idx += 256) {
      const int r = idx / 130, c = idx % 130;
      if (r != p) aug[r][c] -= fac[r] * aug[p][c];
    }
    __syncthreads();
  }

  // A2 = X^T y
  for (int idx = threadIdx.x; idx < 65 * 64; idx += 256) {
    const int i = idx / 64, j = idx % 64;
    float s = 0.0f;
    for (int n = 0; n < 256; ++n) {
      const float xi = (i == 0) ? 1.0f : F[n * 64 + i - 1];
      s += xi * T[n * 64 + j];
    }
    A2[i][j] = s;
  }
  __syncthreads();

  // M = inv * A2  -> stored in left half of aug
  for (int idx = threadIdx.x; idx < 65 * 64; idx += 256) {
    const int i = idx / 64, j = idx % 64;
    float s = 0.0f;
    for (int k = 0; k < 65; ++k) s += aug[i][65 + k] * A2[k][j];
    aug[i][j] = s;
  }
  __syncthreads();

  // resid = targ - X*M ; loss = sum(resid^2)
  float lsum = 0.0f;
  for (int idx = threadIdx.x; idx < 256 * 64; idx += 256) {
    const int n = idx >> 6, j = idx & 63;
    float p4 = aug[0][j];
    for (int i = 1; i <= 64; ++i) p4 += F[n * 64 + i - 1] * aug[i][j];
    const float r = T[n * 64 + j] - p4;
    resid[(size_t)b * 256 * 64 + idx] = r;
    lsum += r * r;
  }
  red[threadIdx.x] = lsum;
  __syncthreads();
  for (int off = 128; off > 0; off >>= 1) {
    if (threadIdx.x < off) red[threadIdx.x] += red[threadIdx.x + off];
    __syncthreads();
  }
  if (threadIdx.x == 0) loss[b] = red[0];
}

// ---------------- Renyi MI (alpha=2): Gram via WMMA, H = 16 - log2(||K||_F^2)
__global__ __launch_bounds__(256) void k_renyi_mi(const float* __restrict__ aptr,
                                                  const float* __restrict__ bptr,
                                                  float* __restrict__ outv) {
  __shared__ float na[256], nb[256];
  __shared__ float sums[3];
  const int b = blockIdx.x;
  const float* A  = aptr + (size_t)b * 256 * 64;
  const float* Bm = bptr + (size_t)b * 256 * 64;
  {
    const int t = threadIdx.x;
    float s1 = 0.0f, s2 = 0.0f;
    for (int d = 0; d < 64; ++d) {
      const float v = A[t * 64 + d];  s1 += v * v;
      const float w = Bm[t * 64 + d]; s2 += w * w;
    }
    na[t] = s1;
    nb[t] = s2;
  }
  if (threadIdx.x < 3) sums[threadIdx.x] = 0.0f;
  __syncthreads();

  const int lane = threadIdx.x & 31, wave = threadIdx.x >> 5;
  const int m = lane & 15, half = lane >> 4;
  float sa = 0.0f, sb = 0.0f, sab = 0.0f;

  for (int t = wave; t < 256; t += 8) {      // 16x16 tiles of the 256x256 Gram
    const int ti = t >> 4, tj = t & 15;
    v8f ga = {}, gb = {};
    for (int k0 = 0; k0 < 64; k0 += 4) {
      const int ka = k0 + 2 * half;
      v2f av, bv;
      av.x = A[(ti * 16 + m) * 64 + ka];
      av.y = A[(ti * 16 + m) * 64 + ka + 1];
      bv.x = A[(tj * 16 + m) * 64 + ka];     // B tile = rows of A^T
      bv.y = A[(tj * 16 + m) * 64 + ka + 1];
      ga = wmma4(av, bv, ga);
      av.x = Bm[(ti * 16 + m) * 64 + ka];
      av.y = Bm[(ti * 16 + m) * 64 + ka + 1];
      bv.x = Bm[(tj * 16 + m) * 64 + ka];
      bv.y = Bm[(tj * 16 + m) * 64 + ka + 1];
      gb = wmma4(av, bv, gb);
    }
#pragma unroll
    for (int r = 0; r < 8; ++r) {
      const int i = ti * 16 + r + 8 * half;
      const int j = tj * 16 + m;
      const float ka_ = expf(-(na[i] + na[j] - 2.0f * ga[r]));  // SIGMA = 1
      const float kb_ = expf(-(nb[i] + nb[j] - 2.0f * gb[r]));
      sa += ka_ * ka_;
      sb += kb_ * kb_;
      const float kab = ka_ * kb_;
      sab += kab * kab;
    }
  }
  atomicAdd(&sums[0], sa);
  atomicAdd(&sums[1], sb);
  atomicAdd(&sums[2], sab);
  __syncthreads();
  if (threadIdx.x == 0) {
    const float Hx  = 16.0f - log2f(sums[0]);   // -log2(S / N^2), N^2 = 2^16
    const float Hy  = 16.0f - log2f(sums[1]);
    const float Hxy = 16.0f - log2f(sums[2]);
    outv[b] = (Hx + Hy - Hxy) / fmaxf(Hx, Hy);
  }
}

// ---------------- head2: BN over batch, relu MLP, softmax
__global__ __launch_bounds__(64) void k_head2(const float* __restrict__ out_c,
                                              const float* __restrict__ ex,
                                              const float* __restrict__ ey,
                                              const float* __restrict__ gamma,
                                              const float* __restrict__ beta,
                                              const float* __restrict__ w2,
                                              const float* __restrict__ b2,
                                              const float* __restrict__ wc2,
                                              const float* __restrict__ bc2,
                                              float* __restrict__ out_cf,
                                              float* __restrict__ prob2) {
  __shared__ float red[64];
  __shared__ float mu[5], var[5];
  const int b = threadIdx.x;
  float inc[5];
  inc[0] = out_c[b * 3 + 0];
  inc[1] = out_c[b * 3 + 1];
  inc[2] = out_c[b * 3 + 2];
  inc[3] = ex[b];
  inc[4] = ey[b];
  for (int f = 0; f < 5; ++f) {
    red[b] = inc[f];
    __syncthreads();
    for (int off = 32; off > 0; off >>= 1) {
      if (b < off) red[b] += red[b + off];
      __syncthreads();
    }
    if (b == 0) mu[f] = red[0] * (1.0f / 64.0f);
    __syncthreads();
    red[b] = inc[f] * inc[f];
    __syncthreads();
    for (int off = 32; off > 0; off >>= 1) {
      if (b < off) red[b] += red[b + off];
      __syncthreads();
    }
    if (b == 0) var[f] = red[0] * (1.0f / 64.0f) - mu[f] * mu[f];
    __syncthreads();
  }
  float v[5];
#pragma unroll
  for (int f = 0; f < 5; ++f)
    v[f] = (inc[f] - mu[f]) * rsqrtf(var[f] + BN_EPS_) * gamma[f] + beta[f];

  float o[3] = {bc2[0], bc2[1], bc2[2]};
  for (int j = 0; j < 64; ++j) {
    float h = b2[j];
#pragma unroll
    for (int f = 0; f < 5; ++f) h += v[f] * w2[f * 64 + j];
    h = fmaxf(h, 0.0f);
#pragma unroll
    for (int k = 0; k < 3; ++k) o[k] += h * wc2[j * 3 + k];
  }
  const float mx = fmaxf(o[0], fmaxf(o[1], o[2]));
  const float e0 = expf(o[0] - mx), e1 = expf(o[1] - mx), e2 = expf(o[2] - mx);
  const float inv = 1.0f / (e0 + e1 + e2);
#pragma unroll
  for (int k = 0; k < 3; ++k) out_cf[b * 3 + k] = o[k];
  prob2[b * 3 + 0] = e0 * inv;
  prob2[b * 3 + 1] = e1 * inv;
  prob2[b * 3 + 2] = e2 * inv;
}

// ---------------------------------------------------------------------------
extern "C" void kernel_launch(void* const* d_in, const int* in_sizes, int n_in,
                              void* d_out, int out_size, void* d_ws, size_t ws_size,
                              hipStream_t stream) {
  (void)in_sizes; (void)n_in; (void)out_size; (void)ws_size;
  const float* x     = (const float*)d_in[0];
  const float* y     = (const float*)d_in[1];
  const float* w11_1 = (const float*)d_in[2];
  const float* b11_1 = (const float*)d_in[3];
  const float* w11_2 = (const float*)d_in[4];
  const float* b11_2 = (const float*)d_in[5];
  const float* w12_1 = (const float*)d_in[6];
  const float* b12_1 = (const float*)d_in[7];
  const float* w12_2 = (const float*)d_in[8];
  const float* b12_2 = (const float*)d_in[9];
  const float* wc1   = (const float*)d_in[10];
  const float* bc1   = (const float*)d_in[11];
  const float* gamma = (const float*)d_in[12];
  const float* beta  = (const float*)d_in[13];
  const float* w2    = (const float*)d_in[14];
  const float* b2    = (const float*)d_in[15];
  const float* wc2   = (const float*)d_in[16];
  const float* bc2   = (const float*)d_in[17];

  float* out = (float*)d_out;
  float* ws  = (float*)d_ws;

  const size_t SL = (size_t)64 * 256 * 64;   // one [B,N,D] tensor = 1,048,576 floats
  float* OUT_X = ws;                 // mlp11(x)
  float* OUT_Y = ws + SL;            // mlp11(y)
  float* H1    = ws + 2 * SL;        // [16384,128] hidden (2*SL); reused below
  float* XX    = H1;                 // fro-normalized x (reuses H1 after mlp12b)
  float* YY    = H1 + SL;            // fro-normalized y
  float* MEAN  = ws + 4 * SL;        // [64,64]
  float* OXN   = MEAN + 64 * 64;     // fro-normalized out_x
  float* OYN   = OXN + SL;           // fro-normalized out_y
  float* RESX  = OYN + SL;           // yy - px
  float* RESY  = RESX + SL;          // xx - py

  float* o_outc  = out;          // [64,3]
  float* o_rx    = out + 192;    // [64]
  float* o_ry    = out + 256;    // [64]
  float* o_ex    = out + 320;    // [64]
  float* o_ey    = out + 384;    // [64]
  float* o_outcf = out + 448;    // [64,3]
  float* o_prob1 = out + 640;    // [64,3]
  float* o_prob2 = out + 832;    // [64,3]

  k_mlp11<<<dim3(1024), dim3(128), 0, stream>>>(x, w11_1, b11_1, w11_2, b11_2, OUT_X);
  k_mlp11<<<dim3(1024), dim3(128), 0, stream>>>(y, w11_1, b11_1, w11_2, b11_2, OUT_Y);
  k_mlp12a<<<dim3(1024), dim3(256), 0, stream>>>(OUT_X, OUT_Y, w12_1, b12_1, H1);
  k_mlp12b<<<dim3(64), dim3(256), 0, stream>>>(H1, w12_2, b12_2, MEAN);
  k_head1<<<dim3(1), dim3(64), 0, stream>>>(MEAN, wc1, bc1, o_outc, o_prob1);

  k_fronorm<<<dim3(64), dim3(256), 0, stream>>>(x, XX);
  k_fronorm<<<dim3(64), dim3(256), 0, stream>>>(y, YY);
  k_fronorm<<<dim3(64), dim3(256), 0, stream>>>(OUT_X, OXN);
  k_fronorm<<<dim3(64), dim3(256), 0, stream>>>(OUT_Y, OYN);

  k_regress<<<dim3(64), dim3(256), 0, stream>>>(OXN, YY, RESX, o_rx);
  k_regress<<<dim3(64), dim3(256), 0, stream>>>(OYN, XX, RESY, o_ry);

  k_renyi_mi<<<dim3(64), dim3(256), 0, stream>>>(RESX, x, o_ex);
  k_renyi_mi<<<dim3(64), dim3(256), 0, stream>>>(RESY, y, o_ey);

  k_head2<<<dim3(1), dim3(64), 0, stream>>>(o_outc, o_ex, o_ey, gamma, beta,
                                            w2, b2, wc2, bc2, o_outcf, o_prob2);
}